// SpikeTruncatedMixtureModel_31945966747651
// MI455X (gfx1250) — compile-verified
//
#include <hip/hip_runtime.h>

#define C_TOP 3
#define S_NB  2
#define NCOL  10      // C + C*S + E
#define NB    64      // N_NEIGHB
#define UMAX  400     // N_UNITS (static LDS sizing)
#define BLK   256

#define GLOBAL_AS __attribute__((address_space(1)))
#define LDS_AS    __attribute__((address_space(3)))

#if defined(__has_builtin)
# if __has_builtin(__builtin_amdgcn_global_load_async_to_lds_b32)
#  define HAVE_ASYNC_B32 1
# endif
# if __has_builtin(__builtin_amdgcn_s_wait_asynccnt)
#  define HAVE_WAIT_ASYNC 1
# endif
#endif

// ---------------- threefry-2x32 (exact JAX semantics, key = (0,1)) -------------
__device__ __forceinline__ unsigned rotl32(unsigned x, int d) {
  return (x << d) | (x >> (32 - d));
}

__device__ __forceinline__ void threefry2x32(unsigned k0, unsigned k1,
                                             unsigned x0, unsigned x1,
                                             unsigned& o0, unsigned& o1) {
  const unsigned ks2 = k0 ^ k1 ^ 0x1BD11BDAu;
  const int R0[4] = {13, 15, 26, 6};
  const int R1[4] = {17, 29, 16, 24};
  x0 += k0; x1 += k1;
#pragma unroll
  for (int r = 0; r < 4; ++r) { x0 += x1; x1 = rotl32(x1, R0[r]); x1 ^= x0; }
  x0 += k1; x1 += ks2 + 1u;
#pragma unroll
  for (int r = 0; r < 4; ++r) { x0 += x1; x1 = rotl32(x1, R1[r]); x1 ^= x0; }
  x0 += ks2; x1 += k0 + 2u;
#pragma unroll
  for (int r = 0; r < 4; ++r) { x0 += x1; x1 = rotl32(x1, R0[r]); x1 ^= x0; }
  x0 += k0; x1 += k1 + 3u;
#pragma unroll
  for (int r = 0; r < 4; ++r) { x0 += x1; x1 = rotl32(x1, R1[r]); x1 ^= x0; }
  x0 += k1; x1 += ks2 + 4u;
#pragma unroll
  for (int r = 0; r < 4; ++r) { x0 += x1; x1 = rotl32(x1, R0[r]); x1 ^= x0; }
  o0 = x0 + ks2;
  o1 = x1 + k0 + 5u;
}

// ---------------- phase 0: zero the (U x NB) counts table ---------------------
__global__ void __launch_bounds__(BLK) zero_k(int* __restrict__ counts, int m) {
  int i = blockIdx.x * BLK + threadIdx.x;
  if (i < m) counts[i] = 0;
}

// ---------------- phase 1: histogram of (top0, nb) ----------------------------
__global__ void __launch_bounds__(BLK) hist_k(const int* __restrict__ top,
                                              const int* __restrict__ nbid,
                                              int* __restrict__ counts,
                                              int n) {
  long long i = (long long)blockIdx.x * BLK + threadIdx.x;
  if (i < n) {
    int t0 = top[3 * i];
    int nb = nbid[i];
    atomicAdd(&counts[t0 * NB + nb], 1);
  }
}

// ---------------- phase 2: per-neighborhood sorted overlap lists --------------
// explore[b*U + k] = ascending unit ids with counts[u][b] > 0, padded with U.
__global__ void build_explore_k(const int* __restrict__ counts,
                                int* __restrict__ explore,
                                int* __restrict__ nexp, int U) {
  int b = threadIdx.x;          // one thread per neighborhood (64)
  if (b >= NB) return;
  int cnt = 0;
  for (int u = 0; u < U; ++u)
    if (counts[u * NB + b] > 0) explore[b * U + cnt++] = u;
  nexp[b] = cnt;
  for (int k = cnt; k < U; ++k) explore[b * U + k] = U;
}

// ---------------- phase 3: per-spike candidate proposal -----------------------
__global__ void __launch_bounds__(BLK)
propose_main(const float* __restrict__ logl,
             const int*   __restrict__ top,
             const int*   __restrict__ usn,
             const int*   __restrict__ nbid,
             int*         __restrict__ counts,
             const int*   __restrict__ explore,
             const int*   __restrict__ nexp,
             float*       __restrict__ out_cand,
             float*       __restrict__ out_scor,
             int n, int U) {
  __shared__ int   s_usn[UMAX * S_NB];
  __shared__ float s_logl[UMAX];
  __shared__ int   s_nexp[NB];
  __shared__ float s_oc[BLK * NCOL];
  __shared__ float s_os[BLK * NCOL];

  const int tid = threadIdx.x;
  const int nUsn = U * S_NB;

  // Stage lookup tables into LDS (CDNA5 async global->LDS path when available)
#if defined(HAVE_ASYNC_B32)
  for (int k = tid; k < nUsn; k += BLK)
    __builtin_amdgcn_global_load_async_to_lds_b32(
        (GLOBAL_AS int*)(usn + k), (LDS_AS int*)(&s_usn[k]), 0, 0);
  for (int k = tid; k < U; k += BLK)
    __builtin_amdgcn_global_load_async_to_lds_b32(
        (GLOBAL_AS int*)(logl + k), (LDS_AS int*)(&s_logl[k]), 0, 0);
  if (tid < NB)
    __builtin_amdgcn_global_load_async_to_lds_b32(
        (GLOBAL_AS int*)(nexp + tid), (LDS_AS int*)(&s_nexp[tid]), 0, 0);
# if defined(HAVE_WAIT_ASYNC)
  __builtin_amdgcn_s_wait_asynccnt(0);
# endif
#else
  for (int k = tid; k < nUsn; k += BLK) s_usn[k] = usn[k];
  for (int k = tid; k < U; k += BLK)    s_logl[k] = logl[k];
  if (tid < NB)                         s_nexp[tid] = nexp[tid];
#endif
  __syncthreads();

  const long long i = (long long)blockIdx.x * BLK + tid;
  const bool active = (i < n);

  if (active) {
    const int* tr = top + 3 * i;
    const int t0 = tr[0], t1 = tr[1], t2 = tr[2];
    const int nb = nbid[i];
    __builtin_prefetch(explore + (size_t)nb * U, 0, 1);  // global_prefetch_b8

    int cand[NCOL];
    cand[0] = t0; cand[1] = t1; cand[2] = t2;
    cand[3] = s_usn[t0 * 2 + 0]; cand[4] = s_usn[t0 * 2 + 1];
    cand[5] = s_usn[t1 * 2 + 0]; cand[6] = s_usn[t1 * 2 + 1];
    cand[7] = s_usn[t2 * 2 + 0]; cand[8] = s_usn[t2 * 2 + 1];

    // JAX uniform(key(1), (n,1)): threefry over counter split [0,h) | [h,2h)
    const unsigned h = (unsigned)(n >> 1);
    const unsigned ii = (unsigned)i;
    unsigned x0, x1, o0, o1;
    if (ii < h) { x0 = ii; x1 = ii + h; } else { x0 = ii - h; x1 = ii; }
    threefry2x32(0u, 1u, x0, x1, o0, o1);
    const unsigned bits = (ii < h) ? o0 : o1;
    float u = __uint_as_float((bits >> 9) | 0x3f800000u) - 1.0f;
    u = fmaxf(u, 0.0f);

    const int ne = s_nexp[nb];
    int ex = -1;
    if (ne > 0) {
      int targ = (int)floorf(u * (float)ne);
      targ = min(targ, ne - 1);
      ex = explore[(size_t)nb * U + targ];
    }
    cand[9] = ex;

    // counts update uses PRE-dedup tail (reference order)
#pragma unroll
    for (int j = 1; j < NCOL; ++j)
      if (cand[j] >= 0) atomicAdd(&counts[cand[j] * NB + nb], 1);

    // dedup (computed against original values) + scores, stage rows in LDS
#pragma unroll
    for (int j = 0; j < NCOL; ++j) {
      bool dup = false;
#pragma unroll
      for (int k = 0; k < NCOL; ++k)
        if (k < j) dup |= (cand[j] == cand[k]);
      const int cj = dup ? -1 : cand[j];
      s_oc[tid * NCOL + j] = (float)cj;
      s_os[tid * NCOL + j] = (cj >= 0) ? s_logl[cj] : 0.0f;
    }
  }
  __syncthreads();

  // Coalesced flush: lane-contiguous dword stores of the block's 2560-elem chunk
  const long long chunk = (long long)blockIdx.x * (BLK * NCOL);
  const long long lim = (long long)n * NCOL;
#pragma unroll
  for (int k = 0; k < NCOL; ++k) {
    const long long e = chunk + (long long)k * BLK + tid;
    if (e < lim) {
      out_cand[e] = s_oc[k * BLK + tid];
      out_scor[e] = s_os[k * BLK + tid];
    }
  }
}

// ---------------- phase 4: counts -> float output -----------------------------
__global__ void __launch_bounds__(BLK) copy_counts_k(const int* __restrict__ counts,
                                                     float* __restrict__ out, int m) {
  int i = blockIdx.x * BLK + threadIdx.x;
  if (i < m) out[i] = (float)counts[i];
}

extern "C" void kernel_launch(void* const* d_in, const int* in_sizes, int n_in,
                              void* d_out, int out_size, void* d_ws, size_t ws_size,
                              hipStream_t stream) {
  const float* logl = (const float*)d_in[0];   // (U,)      f32
  const int*   top  = (const int*)d_in[1];     // (n,3)     i32
  const int*   usn  = (const int*)d_in[2];     // (U,2)     i32
  const int*   nbid = (const int*)d_in[3];     // (n,)      i32

  const int U = in_sizes[0];
  const int n = in_sizes[1] / C_TOP;

  // workspace: counts (U*NB i32) | explore (NB*U i32) | nexp (NB i32)
  int* counts  = (int*)d_ws;
  int* explore = counts + (size_t)U * NB;
  int* nexp    = explore + (size_t)NB * U;

  // d_out: candidates (n*10) | counts (U*NB) | scores (n*10), all as f32
  float* out_cand   = (float*)d_out;
  float* out_counts = out_cand + (long long)n * NCOL;
  float* out_scor   = out_counts + (long long)U * NB;

  const int cblk = (U * NB + BLK - 1) / BLK;
  const int nblk = (int)((n + BLK - 1) / BLK);

  zero_k<<<cblk, BLK, 0, stream>>>(counts, U * NB);
  hist_k<<<nblk, BLK, 0, stream>>>(top, nbid, counts, n);
  build_explore_k<<<1, NB, 0, stream>>>(counts, explore, nexp, U);
  propose_main<<<nblk, BLK, 0, stream>>>(logl, top, usn, nbid, counts, explore,
                                         nexp, out_cand, out_scor, n, U);
  copy_counts_k<<<cblk, BLK, 0, stream>>>(counts, out_counts, U * NB);
}